// NeuSRenderer_77154792506113
// MI455X (gfx1250) — compile-verified
//
#include <hip/hip_runtime.h>

typedef float v2f __attribute__((ext_vector_type(2)));
typedef float v8f __attribute__((ext_vector_type(8)));

namespace {
constexpr int Bq = 2, Dq = 32, Hq = 320, Wq = 960;
constexpr int HW   = Hq * Wq;        // 307200
constexpr int NPIX = Bq * HW;        // 614400
constexpr float FXB = 0.58f * 1242.0f * 0.54f;  // fx * BASELINE = 388.9944
constexpr int ROWP = 36;             // padded LDS row stride: conflict-free WMMA feeds
constexpr size_t O_COLOR = 0;
constexpr size_t O_WSUM  = (size_t)Bq * 3 * HW;        // 1843200
constexpr size_t O_DEPTH = O_WSUM + (size_t)Bq * HW;   // 2457600
constexpr size_t O_WARP  = O_DEPTH + (size_t)Bq * HW;  // 3072000

// hypo[k] = fx*baseline / flip([0.01,1,2,...,31])[k]
constexpr float hyc(int k) {
    return FXB / ((k == 31) ? 0.01f : (float)(31 - k));
}
// Front-end-evaluated table: kernel IR only ever sees float literals, so no
// runtime fdiv can be reconstituted by select-sinking.
constexpr float HYT[32] = {
    hyc(0),  hyc(1),  hyc(2),  hyc(3),  hyc(4),  hyc(5),  hyc(6),  hyc(7),
    hyc(8),  hyc(9),  hyc(10), hyc(11), hyc(12), hyc(13), hyc(14), hyc(15),
    hyc(16), hyc(17), hyc(18), hyc(19), hyc(20), hyc(21), hyc(22), hyc(23),
    hyc(24), hyc(25), hyc(26), hyc(27), hyc(28), hyc(29), hyc(30), hyc(31)
};
}

__device__ __forceinline__ float fast_sigmoid(float x) {
    return __fdividef(1.0f, 1.0f + __expf(-x));
}

__global__ __launch_bounds__(256) void neus_fused(
    const float* __restrict__ sdf, const float* __restrict__ colg,
    const float* __restrict__ right, const float* __restrict__ varp,
    float* __restrict__ out)
{
    __shared__ float wlds[256 * ROWP];   // per-thread weight rows (padded)
    __shared__ float dlds[256];          // depth  (from WMMA)
    __shared__ float slds[256];          // wsum   (from WMMA)

    const int tid = threadIdx.x;
    const int p   = blockIdx.x * 256 + tid;   // pixel id, grid exactly covers NPIX
    const int b   = p / HW;
    const int rem = p - b * HW;
    const int h   = rem / Wq;
    const int w   = rem - h * Wq;

    const float inv_s = __expf(varp[0] * 10.0f);

    const float* sp = sdf  + (size_t)b * Dq * HW + rem;
    const float* cp = colg + (size_t)b * 3 * Dq * HW + rem;

    // sdf_flipped[j] = sdf_grid plane (31-j); stream-once -> non-temporal
    float cprev = fast_sigmoid(__builtin_nontemporal_load(sp + 31 * HW) * inv_s);
    float trans = 1.0f;
    float c0 = 0.0f, c1 = 0.0f, c2 = 0.0f;

    #pragma unroll
    for (int j = 0; j < 31; ++j) {
        const int dd = 30 - j;   // sdf plane for sample j+1
        const float sv = __builtin_nontemporal_load(sp + dd * HW);
        const float cnext = fast_sigmoid(sv * inv_s);
        float alpha = __fdividef(cprev - cnext + 1e-5f, cprev + 1e-5f);
        alpha = fminf(fmaxf(alpha, 0.0f), 1.0f);
        const float wgt = alpha * trans;
        trans *= (1.0f - alpha + 1e-7f);
        const float* cpd = cp + (size_t)(dd + 1) * HW;   // color plane for flip-index j
        c0 = fmaf(__builtin_nontemporal_load(cpd),                      wgt, c0);
        c1 = fmaf(__builtin_nontemporal_load(cpd + (size_t)Dq * HW),    wgt, c1);
        c2 = fmaf(__builtin_nontemporal_load(cpd + (size_t)2 * Dq * HW),wgt, c2);
        wlds[tid * ROWP + j] = wgt;
        cprev = cnext;
    }
    wlds[tid * ROWP + 31] = 0.0f;        // pad K to 32
    __syncthreads();

    // depth / weights-sum reduction on the matrix cores:
    // per wave, two chains of 8x V_WMMA_F32_16X16X4_F32 (K accumulates to 32).
    // A = const [hypo ; ones ; 0...] (16x4 per step), B = 4x16 weights slab
    // for 16 pixels. D row M=0 -> depth, M=1 -> wsum, columns = pixels.
    const int lane = tid & 31;
    const int wid  = tid >> 5;
    const int half = lane >> 4;          // lanes 16-31 carry K+2 per ISA layout
    const int lm   = lane & 15;
    const int r0   = (wid * 32 + lm) * ROWP;       // pixels base..base+15
    const int r1   = r0 + 16 * ROWP;               // pixels base+16..base+31
    v8f acc0 = {};
    v8f acc1 = {};
    #pragma unroll
    for (int t = 0; t < 8; ++t) {
        const int k0 = 4 * t + 2 * half;
        // literal-vs-literal selects (v_cndmask); HYT entries are front-end
        // constants, so no divide can appear here.
        const float h0 = half ? HYT[4 * t + 2] : HYT[4 * t];
        const float h1 = half ? HYT[4 * t + 3] : HYT[4 * t + 1];
        const float a0 = (lm == 0) ? h0 : ((lm == 1) ? 1.0f : 0.0f);
        const float a1 = (lm == 0) ? h1 : ((lm == 1) ? 1.0f : 0.0f);
        v2f A  = {a0, a1};
        v2f B0 = {wlds[r0 + k0], wlds[r0 + k0 + 1]};
        v2f B1 = {wlds[r1 + k0], wlds[r1 + k0 + 1]};
        acc0 = __builtin_amdgcn_wmma_f32_16x16x4_f32(false, A, false, B0,
                                                     (short)0, acc0, false, false);
        acc1 = __builtin_amdgcn_wmma_f32_16x16x4_f32(false, A, false, B1,
                                                     (short)0, acc1, false, false);
    }
    if (half == 0) {   // D rows M=0 (depth) and M=1 (wsum) live in lanes 0-15
        dlds[wid * 32 + lm]      = acc0[0];
        slds[wid * 32 + lm]      = acc0[1];
        dlds[wid * 32 + 16 + lm] = acc1[0];
        slds[wid * 32 + 16 + lm] = acc1[1];
    }
    __syncthreads();
    const float depth = dlds[tid];
    const float wsum  = slds[tid];

    // outputs
    out[O_COLOR + (size_t)(b * 3 + 0) * HW + rem] = c0;
    out[O_COLOR + (size_t)(b * 3 + 1) * HW + rem] = c1;
    out[O_COLOR + (size_t)(b * 3 + 2) * HW + rem] = c2;
    out[O_WSUM  + (size_t)p] = wsum;
    out[O_DEPTH + (size_t)p] = depth;

    // disparity warp: gy maps to exactly row h (align_corners), so 1-D lerp in x.
    const float disp = __fdividef(FXB, depth);
    float ix = (float)w - disp * ((float)(Wq - 1) / (float)Wq);
    ix = fminf(fmaxf(ix, 0.0f), (float)(Wq - 1));
    const float x0f = floorf(ix);
    const float wx  = ix - x0f;
    const int x0 = (int)x0f;
    const int x1 = min(x0 + 1, Wq - 1);
    const float* rp = right + (size_t)b * 3 * HW + (size_t)h * Wq;
    #pragma unroll
    for (int c = 0; c < 3; ++c) {
        const float v0 = rp[(size_t)c * HW + x0];
        const float v1 = rp[(size_t)c * HW + x1];
        out[O_WARP + (size_t)(b * 3 + c) * HW + rem] = fmaf(v1 - v0, wx, v0);
    }
}

extern "C" void kernel_launch(void* const* d_in, const int* in_sizes, int n_in,
                              void* d_out, int out_size, void* d_ws, size_t ws_size,
                              hipStream_t stream) {
    const float* sdf   = (const float*)d_in[0];
    const float* colg  = (const float*)d_in[1];
    const float* right = (const float*)d_in[2];
    const float* varp  = (const float*)d_in[3];
    float* out = (float*)d_out;
    (void)in_sizes; (void)n_in; (void)out_size; (void)d_ws; (void)ws_size;
    neus_fused<<<dim3(NPIX / 256), dim3(256), 0, stream>>>(sdf, colg, right, varp, out);
}